// NTM_Head_24146306138467
// MI455X (gfx1250) — compile-verified
//
#include <hip/hip_runtime.h>
#include <math.h>

// ---- problem constants (fixed by the reference) ----
#define BATCH  4096
#define NSLOT  512
#define MWID   64
#define SEQW   8
#define HID    256
#define CTRLO  128
#define PARAM  70            // M + 1 + 1 + 3 + 1
#define REPR   268           // 2*PARAM + 2*M
#define REPRP  272           // padded repr width (mult of 16)
#define INS    73            // (SEQW+1) + M
#define INSP   80            // padded controller input width (mult of 8)
#define CAT2   192           // CTRLO + M
#define LDSTRIDE 68          // padded bank row stride in LDS (16B-aligned rows)

typedef __attribute__((ext_vector_type(2))) float v2f;
typedef __attribute__((ext_vector_type(8))) float v8f;

__device__ __forceinline__ float sigmoidf_(float x){ return 1.f/(1.f+expf(-x)); }
__device__ __forceinline__ float softplusf_(float x){ return fmaxf(x,0.f)+log1pf(expf(-fabsf(x))); }

template<int ACT>
__device__ __forceinline__ float apply_act(float x){
  if constexpr (ACT==1) return x>0.f ? x : 0.01f*x;      // leaky_relu (slope 0.01)
  else if constexpr (ACT==2) return tanhf(x);
  else if constexpr (ACT==3) return 1.f/(1.f+expf(-x));  // sigmoid
  else return x;
}

// ---------------------------------------------------------------------------
// f32 WMMA GEMM, guard-free inner loop (all operands pre-padded):
//   C[B x Nstore] = act(A[B x K] @ W[K x N] + bias)
// One wave per 16x16 tile; K unrolled by 8 with two independent WMMA chains.
// ---------------------------------------------------------------------------
template<int ACT, bool SGUARD>
__global__ __launch_bounds__(32)
void gemm_wmma_f32(const float* __restrict__ A, const float* __restrict__ W,
                   const float* __restrict__ bias, float* __restrict__ C,
                   int K, int N, int Nstore)
{
  const int lane = threadIdx.x;                    // wave32
  const int nloc = lane & 15;
  const int col  = blockIdx.x * 16 + nloc;         // always < N (padded)
  const int row  = blockIdx.y * 16 + nloc;         // BATCH % 16 == 0
  const int koff = (lane >> 4) << 1;               // lanes 0-15 -> K+0/1, 16-31 -> K+2/3

  const float* aP = A + (size_t)row * K + koff;
  const float* wP = W + col + (size_t)koff * N;
  const size_t n1 = (size_t)N, n4 = 4 * (size_t)N, n5 = 5 * (size_t)N, n8 = 8 * (size_t)N;

  v8f acc0 = {}, acc1 = {};
  for (int k0 = 0; k0 < K; k0 += 8) {
    float2 a01 = *(const float2*)(aP);
    float2 a23 = *(const float2*)(aP + 4);
    float b0 = wP[0], b1 = wP[n1];
    float b2 = wP[n4], b3 = wP[n5];
    v2f av0 = {a01.x, a01.y}; v2f bv0 = {b0, b1};
    v2f av1 = {a23.x, a23.y}; v2f bv1 = {b2, b3};
    acc0 = __builtin_amdgcn_wmma_f32_16x16x4_f32(false, av0, false, bv0,
                                                 (short)0, acc0, false, false);
    acc1 = __builtin_amdgcn_wmma_f32_16x16x4_f32(false, av1, false, bv1,
                                                 (short)0, acc1, false, false);
    aP += 8; wP += n8;
  }
  v8f acc = acc0 + acc1;

  const float bs = bias[col];
  const int rbase = blockIdx.y * 16 + ((lane >> 4) << 3); // +8 for upper half-lanes
  if (!SGUARD || col < Nstore) {
    #pragma unroll
    for (int r = 0; r < 8; ++r)
      C[(size_t)(rbase + r) * Nstore + col] = apply_act<ACT>(acc[r] + bs);
  }
}

// ---- pack / pad kernels ----
__global__ void pack_ctrl_in(const float* __restrict__ x, const float* __restrict__ pr,
                             float* __restrict__ o)
{
  int i = blockIdx.x * blockDim.x + threadIdx.x;
  if (i >= BATCH * INSP) return;
  int b = i / INSP, j = i % INSP;
  float v = 0.f;
  if (j < SEQW + 1)      v = x[b * (SEQW + 1) + j];
  else if (j < INS)      v = pr[b * MWID + (j - (SEQW + 1))];
  o[i] = v;
}

__global__ void pack_cat2(const float* __restrict__ co, const float* __restrict__ rd,
                          float* __restrict__ o)
{
  int i = blockIdx.x * blockDim.x + threadIdx.x;
  if (i >= BATCH * CAT2) return;
  int b = i / CAT2, j = i % CAT2;
  o[i] = (j < CTRLO) ? co[b * CTRLO + j] : rd[b * MWID + (j - CTRLO)];
}

__global__ void pad2d(const float* __restrict__ src, float* __restrict__ dst,
                      int srcR, int srcC, int dstR, int dstC)
{
  int i = blockIdx.x * blockDim.x + threadIdx.x;
  if (i >= dstR * dstC) return;
  int r = i / dstC, c = i % dstC;
  dst[i] = (r < srcR && c < srcC) ? src[(size_t)r * srcC + c] : 0.f;
}

// ---- workgroup reductions over 256 threads ----
__device__ __forceinline__ float blk_max(float v, float* red){
  int t = threadIdx.x;
  red[t] = v; __syncthreads();
  #pragma unroll
  for (int s = 128; s > 0; s >>= 1){
    if (t < s) red[t] = fmaxf(red[t], red[t + s]);
    __syncthreads();
  }
  float r = red[0]; __syncthreads(); return r;
}
__device__ __forceinline__ float blk_sum(float v, float* red){
  int t = threadIdx.x;
  red[t] = v; __syncthreads();
  #pragma unroll
  for (int s = 128; s > 0; s >>= 1){
    if (t < s) red[t] = red[t] + red[t + s];
    __syncthreads();
  }
  float r = red[0]; __syncthreads(); return r;
}

// ---------------------------------------------------------------------------
// NTM head: one workgroup (256 threads) per batch. Bank[b] (128 KB) staged in
// LDS via CDNA5 async global->LDS loads; all addressing math + read + update
// done with exactly one HBM read and one HBM write of the bank.
// ---------------------------------------------------------------------------
__global__ __launch_bounds__(256)
void ntm_head_kernel(const float* __restrict__ reprx,
                     const float* __restrict__ bank,
                     const float* __restrict__ w_read,
                     const float* __restrict__ w_write,
                     float* __restrict__ readv,
                     float* __restrict__ out_bank)
{
  extern __shared__ float smem[];
  float* S_bank = smem;                               // NSLOT*LDSTRIDE
  float* S_ktR  = S_bank + NSLOT * LDSTRIDE;          // 64
  float* S_ktW  = S_ktR + MWID;                       // 64
  float* S_amE  = S_ktW + MWID;                       // 64  (a - e)
  float* S_cosR = S_amE + MWID;                       // 512 (cos -> final read weight)
  float* S_cosW = S_cosR + NSLOT;                     // 512 (cos -> final write weight)
  float* S_wgR  = S_cosW + NSLOT;                     // 512
  float* S_wgW  = S_wgR + NSLOT;                      // 512
  float* S_red  = S_wgW + NSLOT;                      // 256

  const int b = blockIdx.x;
  const int t = threadIdx.x;
  const float* rp    = reprx + (size_t)b * REPRP;
  const float* bankG = bank  + (size_t)b * NSLOT * MWID;

  // ---- stage bank[b] into LDS (async global->LDS, 16B per lane per issue) ----
  #pragma unroll
  for (int i = 0; i < 32; ++i) {
    int f = t + i * 256;                    // 8192 float4
    int e = f * 4, n = e >> 6, m = e & 63;
    unsigned lds_off = (unsigned)(uintptr_t)(S_bank + n * LDSTRIDE + m);
    const float* ga = bankG + e;
    asm volatile("global_load_async_to_lds_b128 %0, %1, off"
                 :: "v"(lds_off), "v"(ga) : "memory");
  }
  if (t < MWID) {
    S_ktR[t] = tanhf(rp[t]);
    S_ktW[t] = tanhf(rp[PARAM + t]);
    S_amE[t] = tanhf(rp[2 * PARAM + MWID + t]) - sigmoidf_(rp[2 * PARAM + t]);
  }
  asm volatile("s_wait_asynccnt 0x0" ::: "memory");
  __syncthreads();

  // key norms (cheap; redundant per thread, broadcast LDS reads)
  float nktR = 0.f, nktW = 0.f;
  for (int m = 0; m < MWID; ++m) { nktR += S_ktR[m] * S_ktR[m]; nktW += S_ktW[m] * S_ktW[m]; }
  nktR = sqrtf(nktR); nktW = sqrtf(nktW);

  // ---- pass 1: cosine similarities (2 rows / thread) ----
  for (int rep = 0; rep < 2; ++rep) {
    int n = t + rep * 256;
    const float* row = S_bank + n * LDSTRIDE;
    float dr = 0.f, dw = 0.f, ss = 0.f;
    for (int m = 0; m < MWID; ++m) {
      float bv = row[m];
      dr += bv * S_ktR[m]; dw += bv * S_ktW[m]; ss += bv * bv;
    }
    float nb = sqrtf(ss);
    S_cosR[n] = dr / fmaxf(nb * nktR, 1e-8f);
    S_cosW[n] = dw / fmaxf(nb * nktW, 1e-8f);
  }
  __syncthreads();

  // ---- addressing for both heads ----
  for (int head = 0; head < 2; ++head) {
    float* Scos = head ? S_cosW : S_cosR;
    float* Swg  = head ? S_wgW  : S_wgR;
    const float* wprev = (head ? w_write : w_read) + (size_t)b * NSLOT;
    const float beta  = softplusf_(rp[head * PARAM + MWID]);
    const float gg    = sigmoidf_(rp[head * PARAM + MWID + 1]);
    float s0 = rp[head * PARAM + MWID + 2];
    float s1 = rp[head * PARAM + MWID + 3];
    float s2 = rp[head * PARAM + MWID + 4];
    const float gamma = softplusf_(rp[head * PARAM + MWID + 5]) + 1.f;
    // softmax over the 3 shift logits
    float smx = fmaxf(s0, fmaxf(s1, s2));
    s0 = expf(s0 - smx); s1 = expf(s1 - smx); s2 = expf(s2 - smx);
    float sinv = 1.f / (s0 + s1 + s2);
    s0 *= sinv; s1 *= sinv; s2 *= sinv;

    // content softmax over N=512
    float z0 = beta * Scos[t], z1 = beta * Scos[t + 256];
    float mx  = blk_max(fmaxf(z0, z1), S_red);
    float e0  = expf(z0 - mx), e1 = expf(z1 - mx);
    float sum = blk_sum(e0 + e1, S_red);
    float inv = 1.f / sum;
    Swg[t]       = gg * (e0 * inv) + (1.f - gg) * wprev[t];
    Swg[t + 256] = gg * (e1 * inv) + (1.f - gg) * wprev[t + 256];
    __syncthreads();

    // circular shift (-1,0,+1) then sharpen + renormalize
    float pv0, pv1;
    {
      int n = t;
      float wsv = Swg[(n + NSLOT - 1) & (NSLOT - 1)] * s0 + Swg[n] * s1
                + Swg[(n + 1) & (NSLOT - 1)] * s2;
      pv0 = powf(wsv, gamma);
      n = t + 256;
      wsv = Swg[(n + NSLOT - 1) & (NSLOT - 1)] * s0 + Swg[n] * s1
          + Swg[(n + 1) & (NSLOT - 1)] * s2;
      pv1 = powf(wsv, gamma);
    }
    float tot  = blk_sum(pv0 + pv1, S_red);
    float itot = 1.f / tot;
    Scos[t]       = pv0 * itot;   // final head weight overwrites cos storage
    Scos[t + 256] = pv1 * itot;
    __syncthreads();
  }

  // ---- memory read: read[m] = sum_n wR[n] * bank[n,m] ----
  {
    int m = t & 63, ch = t >> 6;            // 4 n-chunks of 128
    float p = 0.f;
    for (int i = 0; i < NSLOT / 4; ++i) {
      int n = ch * (NSLOT / 4) + i;
      p += S_cosR[n] * S_bank[n * LDSTRIDE + m];
    }
    S_red[t] = p;
    __syncthreads();
    if (t < MWID)
      readv[(size_t)b * MWID + t] = S_red[t] + S_red[t + 64] + S_red[t + 128] + S_red[t + 192];
  }

  // ---- erase/add update, stream straight to HBM (aligned 16B LDS reads) ----
  float4* ob4 = (float4*)(out_bank + (size_t)b * NSLOT * MWID);
  #pragma unroll
  for (int i = 0; i < 32; ++i) {
    int f = t + i * 256;
    int e = f * 4, n = e >> 6, m = e & 63;
    float4 sv = *(const float4*)(S_bank + n * LDSTRIDE + m);
    float w = S_cosW[n];
    float4 v;
    v.x = sv.x + w * S_amE[m + 0];
    v.y = sv.y + w * S_amE[m + 1];
    v.z = sv.z + w * S_amE[m + 2];
    v.w = sv.w + w * S_amE[m + 3];
    ob4[f] = v;
  }
}

// ---------------------------------------------------------------------------
extern "C" void kernel_launch(void* const* d_in, const int* in_sizes, int n_in,
                              void* d_out, int out_size, void* d_ws, size_t ws_size,
                              hipStream_t stream)
{
  const float* x       = (const float*)d_in[0];
  const float* prevr   = (const float*)d_in[1];
  const float* bank    = (const float*)d_in[2];
  const float* w_read  = (const float*)d_in[3];
  const float* w_write = (const float*)d_in[4];
  const float* W0   = (const float*)d_in[5];
  const float* b0   = (const float*)d_in[6];
  const float* W1   = (const float*)d_in[7];
  const float* b1   = (const float*)d_in[8];
  const float* Wout = (const float*)d_in[9];
  const float* bout = (const float*)d_in[10];
  const float* Wr   = (const float*)d_in[11];
  const float* br   = (const float*)d_in[12];
  const float* Wo   = (const float*)d_in[13];
  const float* bo   = (const float*)d_in[14];

  float* out      = (float*)d_out;
  float* out_bank = out + (size_t)BATCH * SEQW;   // outputs concatenated: seq_out, new_bank

  float* ws = (float*)d_ws;
  float* ctrl_in  = ws; ws += (size_t)BATCH * INSP;
  float* h0       = ws; ws += (size_t)BATCH * HID;
  float* h1       = ws; ws += (size_t)BATCH * HID;
  float* ctrl_out = ws; ws += (size_t)BATCH * CTRLO;
  float* reprx    = ws; ws += (size_t)BATCH * REPRP;
  float* readv    = ws; ws += (size_t)BATCH * MWID;
  float* cat2     = ws; ws += (size_t)BATCH * CAT2;
  float* W0p      = ws; ws += (size_t)INSP * HID;     // 80x256
  float* Wrp      = ws; ws += (size_t)CTRLO * REPRP;  // 128x272
  float* brp      = ws; ws += REPRP;                  // 272
  float* Wop      = ws; ws += (size_t)CAT2 * 16;      // 192x16
  float* bop      = ws; ws += 16;                     // 16

  const int SMEM_BYTES = (NSLOT * LDSTRIDE + 3 * MWID + 4 * NSLOT + 256) * (int)sizeof(float);
  hipFuncSetAttribute((const void*)ntm_head_kernel,
                      hipFuncAttributeMaxDynamicSharedMemorySize, SMEM_BYTES);

  // zero-pad weights so every GEMM runs guard-free
  pad2d<<<(INSP * HID   + 255) / 256, 256, 0, stream>>>(W0, W0p, INS,   HID,  INSP,  HID);
  pad2d<<<(CTRLO * REPRP+ 255) / 256, 256, 0, stream>>>(Wr, Wrp, CTRLO, REPR, CTRLO, REPRP);
  pad2d<<<(REPRP        + 255) / 256, 256, 0, stream>>>(br, brp, 1,     REPR, 1,     REPRP);
  pad2d<<<(CAT2 * 16    + 255) / 256, 256, 0, stream>>>(Wo, Wop, CAT2,  SEQW, CAT2,  16);
  pad2d<<<(16           + 255) / 256, 256, 0, stream>>>(bo, bop, 1,     SEQW, 1,     16);

  // controller MLP (WMMA f32 GEMM chain, no inner-loop guards)
  pack_ctrl_in<<<(BATCH * INSP + 255) / 256, 256, 0, stream>>>(x, prevr, ctrl_in);
  gemm_wmma_f32<1,false><<<dim3(HID / 16,  BATCH / 16), 32, 0, stream>>>(ctrl_in, W0p,  b0,   h0,       INSP, HID,   HID);
  gemm_wmma_f32<1,false><<<dim3(HID / 16,  BATCH / 16), 32, 0, stream>>>(h0,      W1,   b1,   h1,       HID,  HID,   HID);
  gemm_wmma_f32<2,false><<<dim3(CTRLO / 16,BATCH / 16), 32, 0, stream>>>(h1,      Wout, bout, ctrl_out, HID,  CTRLO, CTRLO);
  gemm_wmma_f32<0,false><<<dim3(REPRP / 16,BATCH / 16), 32, 0, stream>>>(ctrl_out,Wrp,  brp,  reprx,    CTRLO,REPRP, REPRP);

  // addressing + read + bank update (single HBM pass over bank)
  ntm_head_kernel<<<BATCH, 256, SMEM_BYTES, stream>>>(reprx, bank, w_read, w_write,
                                                      readv, out_bank);

  // output projection (store guarded to the 8 real columns)
  pack_cat2<<<(BATCH * CAT2 + 255) / 256, 256, 0, stream>>>(ctrl_out, readv, cat2);
  gemm_wmma_f32<3,true><<<dim3(1, BATCH / 16), 32, 0, stream>>>(cat2, Wop, bop, out, CAT2, 16, SEQW);
}